// SparseNoisyMoE_2061584302701
// MI455X (gfx1250) — compile-verified
//
#include <hip/hip_runtime.h>
#include <hip/hip_bf16.h>

// Problem constants (match reference)
constexpr int NB   = 8192;  // batch
constexpr int CIN  = 512;   // in features
constexpr int COUT = 720;   // out features
constexpr int NE   = 16;    // experts
// K (top-k) = 2

typedef __attribute__((ext_vector_type(2))) float v2f;
typedef __attribute__((ext_vector_type(8))) float v8f;

// ---------------------------------------------------------------------------
// Kernel 1: gating. One wave32 per row.
//   lane = (expert e = lane&15, half h = lane>>4); each lane dots half the row,
//   xor-16 shuffle combines halves. Top-2 + softmax computed redundantly in all
//   lanes; lane 0 does the bucket scatter + LDS stat accumulation.
// ---------------------------------------------------------------------------
__global__ __launch_bounds__(256) void moe_gate_kernel(
    const float* __restrict__ x, const float* __restrict__ Wg,
    const float* __restrict__ bg,
    int* __restrict__ counts, float* __restrict__ psum, float* __restrict__ dcount,
    int* __restrict__ rowbuf, float* __restrict__ wbuf)
{
    __shared__ float psum_s[NE];
    __shared__ float dcount_s[NE];
    if (threadIdx.x < NE) { psum_s[threadIdx.x] = 0.0f; dcount_s[threadIdx.x] = 0.0f; }
    __syncthreads();

    const int lane = threadIdx.x & 31;
    const int wave = threadIdx.x >> 5;
    const int row  = blockIdx.x * 8 + wave;

    const int e = lane & 15;
    const int h = lane >> 4;

    // partial dot: half-row h for expert e
    float acc = 0.0f;
    const float* xr = x + (size_t)row * CIN;
    for (int k = h * (CIN / 2); k < (h + 1) * (CIN / 2); ++k)
        acc = fmaf(xr[k], Wg[k * NE + e], acc);
    acc += __shfl_xor(acc, 16, 32);          // combine halves; all lanes hold full dot
    float gv = acc + bg[e];                  // gate logit for expert e (MOE_TEMP = 1)

    // gather all 16 logits into every lane (no divergence around shuffles)
    float g[NE];
#pragma unroll
    for (int i = 0; i < NE; ++i) g[i] = __shfl(gv, i, 32);

    // top-2 (first-occurrence tie break, matching jax.lax.top_k)
    int i1 = 0;
#pragma unroll
    for (int i = 1; i < NE; ++i) if (g[i] > g[i1]) i1 = i;
    int i2 = (i1 == 0) ? 1 : 0;
#pragma unroll
    for (int i = 0; i < NE; ++i) if (i != i1 && g[i] > g[i2]) i2 = i;

    // softmax over the two selected logits
    float d  = __expf(g[i2] - g[i1]);
    float w1 = 1.0f / (1.0f + d);
    float w2 = d * w1;

    // full softmax over 16 (for load-balance loss P)
    float mv = g[i1];
    float s = 0.0f;
    float p[NE];
#pragma unroll
    for (int i = 0; i < NE; ++i) { p[i] = __expf(g[i] - mv); s += p[i]; }
    float inv_s = 1.0f / s;

    if (lane == 0) {
        atomicAdd(&dcount_s[i1], 1.0f);
#pragma unroll
        for (int i = 0; i < NE; ++i) atomicAdd(&psum_s[i], p[i] * inv_s);
        int pos1 = atomicAdd(&counts[i1], 1);
        rowbuf[i1 * NB + pos1] = row;
        wbuf  [i1 * NB + pos1] = w1;
        int pos2 = atomicAdd(&counts[i2], 1);
        rowbuf[i2 * NB + pos2] = row;
        wbuf  [i2 * NB + pos2] = w2;
    }

    __syncthreads();
    if (threadIdx.x < NE) {
        atomicAdd(&psum[threadIdx.x],   psum_s[threadIdx.x]);
        atomicAdd(&dcount[threadIdx.x], dcount_s[threadIdx.x]);
    }
}

// ---------------------------------------------------------------------------
// Kernel 2: grouped expert GEMM with f32 WMMA (16x16x4).
//   grid = (COUT/16 N-tiles, NE experts); block = 256 (8 waves).
//   We[e][:, nb:nb+16] staged once per block in LDS (32 KB) -> We read from
//   global exactly once in the whole kernel. Each wave processes 16-row M-tiles
//   of this expert's bucket; K-loop of 128 v_wmma_f32_16x16x4_f32.
//   Output scattered via f32 atomics (exactly 2 contributions per element).
// ---------------------------------------------------------------------------
__global__ __launch_bounds__(256) void moe_expert_gemm_kernel(
    const float* __restrict__ x, const float* __restrict__ We,
    const float* __restrict__ be,
    const int* __restrict__ counts, const int* __restrict__ rowbuf,
    const float* __restrict__ wbuf, float* __restrict__ out)
{
    const int e  = blockIdx.y;
    const int nb = blockIdx.x * 16;

    __shared__ float Bs[CIN][16];   // We slab: [k][n], 32 KB
    __shared__ float bes[16];

    const float* Wep = We + (size_t)e * CIN * COUT + nb;
    for (int idx = threadIdx.x; idx < CIN * 16; idx += 256) {
        int k = idx >> 4, n = idx & 15;
        Bs[k][n] = Wep[(size_t)k * COUT + n];
    }
    if (threadIdx.x < 16) bes[threadIdx.x] = be[e * COUT + nb + threadIdx.x];
    __syncthreads();

    const int cnt = counts[e];
    const int nTiles = (cnt + 15) >> 4;

    const int wave = threadIdx.x >> 5;
    const int lane = threadIdx.x & 31;
    const int m    = lane & 15;     // A-matrix row for this lane
    const int kh   = lane >> 4;     // K-half: lanes 16-31 hold K=2,3 per step

    const int* rb = rowbuf + e * NB;
    const float* wb = wbuf + e * NB;

    for (int t = wave; t < nTiles; t += 8) {
        // A row for this lane (clamped select, no divergence — EXEC stays full)
        int  ridx  = t * 16 + m;
        int  rowm  = rb[ridx < cnt ? ridx : cnt - 1];
        const float* xr = x + (size_t)rowm * CIN;

        v8f acc = {0.f, 0.f, 0.f, 0.f, 0.f, 0.f, 0.f, 0.f};
#pragma unroll 8
        for (int kb = 0; kb < CIN; kb += 4) {
            v2f a = *(const v2f*)(xr + kb + 2 * kh);          // K = kb+2kh, kb+2kh+1
            v2f b;
            b.x = Bs[kb + 2 * kh][m];                         // B[k][n], n = lane&15
            b.y = Bs[kb + 2 * kh + 1][m];
            acc = __builtin_amdgcn_wmma_f32_16x16x4_f32(
                false, a, false, b, (short)0, acc, false, false);
        }

        // D layout: lane holds column n = lane&15, rows mo = v + 8*kh
        const int n = lane & 15;
#pragma unroll
        for (int v = 0; v < 8; ++v) {
            int rj = t * 16 + (v + 8 * kh);
            if (rj < cnt) {
                int   rr = rb[rj];
                float ww = wb[rj];
                atomicAdd(&out[(size_t)rr * COUT + nb + n], ww * (acc[v] + bes[n]));
            }
        }
    }
}

// ---------------------------------------------------------------------------
// Kernel 3: load-balance loss finalize (single wave).
// ---------------------------------------------------------------------------
__global__ void moe_lb_kernel(const float* __restrict__ psum,
                              const float* __restrict__ dcount,
                              float* __restrict__ out_lb)
{
    int l = threadIdx.x;
    float v = 0.0f;
    if (l < NE) {
        float D = dcount[l] / (float)NB;
        float P = psum[l]   / (float)NB;
        v = D * P * (float)NE;
    }
#pragma unroll
    for (int s = 8; s >= 1; s >>= 1) v += __shfl_down(v, s, 32);
    if (l == 0) *out_lb = v;
}

// ---------------------------------------------------------------------------
extern "C" void kernel_launch(void* const* d_in, const int* in_sizes, int n_in,
                              void* d_out, int out_size, void* d_ws, size_t ws_size,
                              hipStream_t stream) {
    const float* x  = (const float*)d_in[0];   // [B, IN]
    const float* Wg = (const float*)d_in[1];   // [IN, E]
    const float* bg = (const float*)d_in[2];   // [E]
    const float* We = (const float*)d_in[3];   // [E, IN, OUT]
    const float* be = (const float*)d_in[4];   // [E, OUT]
    float* out = (float*)d_out;                // [B*OUT] output ++ [1] lb_loss

    // workspace layout
    char*  ws     = (char*)d_ws;
    int*   counts = (int*)   ws;                 // [NE]
    float* psum   = (float*)(ws + 64);           // [NE]
    float* dcount = (float*)(ws + 128);          // [NE]
    int*   rowbuf = (int*)  (ws + 256);                              // [NE*NB]
    float* wbuf   = (float*)(ws + 256 + sizeof(int) * NE * NB);      // [NE*NB]

    hipMemsetAsync(ws, 0, 256, stream);
    hipMemsetAsync(d_out, 0, sizeof(float) * ((size_t)NB * COUT + 1), stream);

    moe_gate_kernel<<<NB / 8, 256, 0, stream>>>(x, Wg, bg, counts, psum, dcount,
                                                rowbuf, wbuf);

    dim3 grid(COUT / 16, NE);
    moe_expert_gemm_kernel<<<grid, 256, 0, stream>>>(x, We, be, counts, rowbuf,
                                                     wbuf, out);

    moe_lb_kernel<<<1, 32, 0, stream>>>(psum, dcount, out + (size_t)NB * COUT);
}